// MGCN_12489764897112
// MI455X (gfx1250) — compile-verified
//
#include <hip/hip_runtime.h>

// MGCN (R-GCN basis decomposition) forward on gfx1250.
//
// Strategy: per layer, precompute Y_b = X @ basis_b (node-level f32 WMMA GEMM,
// avoids any per-edge matmul), then per edge blend 4 Y rows with
// att[type]*norm and scatter with hardware f32 atomics, then finalize with
// mean + X@root + bias (+relu). Memory-bound (~2 GB/layer of edge traffic at
// 23.3 TB/s); GEMMs are ~16 GFLOP/layer so exact-precision
// V_WMMA_F32_16X16X4_F32 is the right matrix op (no need for bf16).

typedef float v2f __attribute__((ext_vector_type(2)));
typedef float v8f __attribute__((ext_vector_type(8)));

// ---------------------------------------------------------------------------
// x[n][j] = emb[entity[n]][j]
__global__ __launch_bounds__(256) void mgcn_gather_kernel(
    const float* __restrict__ emb, const int* __restrict__ ent,
    float* __restrict__ x, int N)
{
    int idx = blockIdx.x * 256 + threadIdx.x;
    if (idx >= N * 128) return;
    int n = idx >> 7, j = idx & 127;
    x[idx] = emb[(size_t)ent[n] * 128 + j];
}

// ---------------------------------------------------------------------------
// C[m][n0..n0+127] = X[m][0..127] @ W[128][128]  (W row-major, k-major)
// One wave -> 16x128 strip via 8 f32 WMMA accumulators; W staged in LDS.
__global__ __launch_bounds__(256) void mgcn_gemm_x_w_kernel(
    const float* __restrict__ X, const float* __restrict__ W,
    float* __restrict__ C, int ldC, int N, int mtiles)
{
    __shared__ float Ws[128 * 128];  // 64 KB of the WGP's 320 KB LDS
    {
        const float4* Wv = (const float4*)W;
        float4* Sv = (float4*)Ws;
        for (int i = threadIdx.x; i < 128 * 128 / 4; i += 256) Sv[i] = Wv[i];
    }
    __syncthreads();

    int wave = threadIdx.x >> 5;
    int lane = threadIdx.x & 31;
    int mt = blockIdx.x * 8 + wave;
    if (mt >= mtiles) return;

    int half = lane >> 4;   // lanes 16-31 carry the K+2/K+3 pair (ISA A layout)
    int l = lane & 15;
    int arow = mt * 16 + l;
    if (arow >= N) arow = N - 1;           // duplicate last row; stores guarded
    const float* Xrow = X + (size_t)arow * 128 + 2 * half;

    v8f acc[8] = {};
    for (int k = 0; k < 128; k += 4) {
        v2f a;
        a.x = Xrow[k];
        a.y = Xrow[k + 1];
        const float* wb = Ws + (k + 2 * half) * 128 + l;
        #pragma unroll
        for (int t = 0; t < 8; ++t) {
            v2f b;
            b.x = wb[t * 16];        // B[k + 2*half    ][n0 + l]
            b.y = wb[t * 16 + 128];  // B[k + 2*half + 1][n0 + l]
            acc[t] = __builtin_amdgcn_wmma_f32_16x16x4_f32(
                false, a, false, b, (short)0, acc[t], false, false);
        }
    }

    // D layout: VGPR r holds row (m0 + half*8 + r), col n0 + (lane&15)
    int row0 = mt * 16 + half * 8;
    #pragma unroll
    for (int t = 0; t < 8; ++t) {
        #pragma unroll
        for (int r = 0; r < 8; ++r) {
            int row = row0 + r;
            if (row < N) C[(size_t)row * ldC + t * 16 + l] = acc[t][r];
        }
    }
}

// ---------------------------------------------------------------------------
// Per edge e (128 lanes): msg[j] = sum_b att[type][b]*norm * Y[src][b][j];
// atomicAdd into agg[dst][j]; lane 0 bumps cnt[dst].
__global__ __launch_bounds__(256) void mgcn_edge_scatter_kernel(
    const float* __restrict__ Y, const int* __restrict__ src,
    const int* __restrict__ dst, const int* __restrict__ etype,
    const float* __restrict__ enorm, const float* __restrict__ att,
    float* __restrict__ agg, float* __restrict__ cnt, int E)
{
    int e = blockIdx.x * 2 + (threadIdx.x >> 7);
    if (e >= E) return;
    int j = threadIdx.x & 127;
    int t = etype[e];
    float nrm = enorm[e];
    int s = src[e];
    int d = dst[e];
    float c0 = att[t * 4 + 0] * nrm;
    float c1 = att[t * 4 + 1] * nrm;
    float c2 = att[t * 4 + 2] * nrm;
    float c3 = att[t * 4 + 3] * nrm;
    const float* yb = Y + (size_t)s * 512;  // node layout [n][b][j], 2KB/node
    float m = c0 * yb[j] + c1 * yb[128 + j] + c2 * yb[256 + j] + c3 * yb[384 + j];
    __hip_atomic_fetch_add(&agg[(size_t)d * 128 + j], m,
                           __ATOMIC_RELAXED, __HIP_MEMORY_SCOPE_AGENT);
    if (j == 0)
        __hip_atomic_fetch_add(&cnt[d], 1.0f,
                               __ATOMIC_RELAXED, __HIP_MEMORY_SCOPE_AGENT);
}

// ---------------------------------------------------------------------------
// out[n][j] = agg[n][j]/max(cnt[n],1) + xroot[n][j] + bias[j]  (+relu)
__global__ __launch_bounds__(256) void mgcn_finalize_kernel(
    const float* __restrict__ agg, const float* __restrict__ cnt,
    const float* __restrict__ xroot, const float* __restrict__ bias,
    float* __restrict__ out, int N, int relu)
{
    int idx = blockIdx.x * 256 + threadIdx.x;
    if (idx >= N * 128) return;
    int n = idx >> 7, j = idx & 127;
    float v = agg[idx] / fmaxf(cnt[n], 1.0f) + xroot[idx] + bias[j];
    if (relu) v = fmaxf(v, 0.0f);
    out[idx] = v;
}

// ---------------------------------------------------------------------------
static void mgcn_run_layer(const float* xin, const float* basis, const float* att,
                           const float* root, const float* bias,
                           const int* src, const int* dst, const int* etype,
                           const float* enorm,
                           float* Y, float* agg, float* cnt, float* out,
                           int N, int E, int relu, hipStream_t stream)
{
    hipMemsetAsync(agg, 0, (size_t)N * 128 * sizeof(float), stream);
    hipMemsetAsync(cnt, 0, (size_t)N * sizeof(float), stream);

    int mtiles = (N + 15) / 16;
    int gblocks = (mtiles + 7) / 8;
    // Y[n][b][j] = (X @ basis_b)[n][j]  -> 4 GEMMs, output col-offset b*128
    for (int b = 0; b < 4; ++b)
        mgcn_gemm_x_w_kernel<<<gblocks, 256, 0, stream>>>(
            xin, basis + (size_t)b * 128 * 128, Y + b * 128, 512, N, mtiles);

    mgcn_edge_scatter_kernel<<<(E + 1) / 2, 256, 0, stream>>>(
        Y, src, dst, etype, enorm, att, agg, cnt, E);

    // Y is dead after the scatter: reuse its first N*128 floats for X @ root.
    float* xroot = Y;
    mgcn_gemm_x_w_kernel<<<gblocks, 256, 0, stream>>>(
        xin, root, xroot, 128, N, mtiles);

    mgcn_finalize_kernel<<<(N * 128 + 255) / 256, 256, 0, stream>>>(
        agg, cnt, xroot, bias, out, N, relu);
}

extern "C" void kernel_launch(void* const* d_in, const int* in_sizes, int n_in,
                              void* d_out, int out_size, void* d_ws, size_t ws_size,
                              hipStream_t stream)
{
    const int*   entity = (const int*)d_in[0];
    const int*   eidx   = (const int*)d_in[1];   // [2, E]: row0=src, row1=dst
    const int*   etype  = (const int*)d_in[2];
    const float* enorm  = (const float*)d_in[3];
    const float* emb    = (const float*)d_in[4];
    const float* basis1 = (const float*)d_in[5];
    const float* att1   = (const float*)d_in[6];
    const float* root1  = (const float*)d_in[7];
    const float* bias1  = (const float*)d_in[8];
    const float* basis2 = (const float*)d_in[9];
    const float* att2   = (const float*)d_in[10];
    const float* root2  = (const float*)d_in[11];
    const float* bias2  = (const float*)d_in[12];

    int N = in_sizes[0];
    int E = in_sizes[2];
    const int* src = eidx;
    const int* dst = eidx + E;

    // Workspace layout (floats): x_cur, x_nxt, agg: N*128 each; cnt: N;
    // Y: N*512 (xroot aliased into Y after the scatter). ~359 MB total.
    float* ws = (float*)d_ws;
    size_t nd = (size_t)N * 128;
    float* x_cur = ws;  ws += nd;
    float* x_nxt = ws;  ws += nd;
    float* agg   = ws;  ws += nd;
    float* cnt   = ws;  ws += N;
    float* Y     = ws;  // N*512

    mgcn_gather_kernel<<<(int)((nd + 255) / 256), 256, 0, stream>>>(
        emb, entity, x_cur, N);

    // conv1, conv1 (+relu), conv2 — exactly as the reference forward.
    mgcn_run_layer(x_cur, basis1, att1, root1, bias1, src, dst, etype, enorm,
                   Y, agg, cnt, x_nxt, N, E, /*relu=*/0, stream);
    mgcn_run_layer(x_nxt, basis1, att1, root1, bias1, src, dst, etype, enorm,
                   Y, agg, cnt, x_cur, N, E, /*relu=*/1, stream);
    mgcn_run_layer(x_cur, basis2, att2, root2, bias2, src, dst, etype, enorm,
                   Y, agg, cnt, (float*)d_out, N, E, /*relu=*/0, stream);
}